// RGCNLayer_50620484550703
// MI455X (gfx1250) — compile-verified
//
#include <hip/hip_runtime.h>

typedef float v2f __attribute__((ext_vector_type(2)));
typedef float v4f __attribute__((ext_vector_type(4)));
typedef float v8f __attribute__((ext_vector_type(8)));

#define D_DIM 128

// ---------------------------------------------------------------------------
// Pass 1: edge scatter.  One wave32 per edge; each lane moves 4 floats.
// agg1[dst] += nodes[src] ; agg2[src] += nodes[dst] ; cnt[dst]++ ; cnt[src]++
// All accumulators live in L2 (51 MB << 192 MB), atomics are f32 HW atomics.
// ---------------------------------------------------------------------------
__global__ void __launch_bounds__(256) rgcn_scatter_kernel(
    const float* __restrict__ nodes, const int* __restrict__ edges,
    float* __restrict__ agg1, float* __restrict__ agg2,
    float* __restrict__ cnt, int E)
{
    int gwave = (int)((blockIdx.x * blockDim.x + threadIdx.x) >> 5);
    int lane  = threadIdx.x & 31;
    if (gwave >= E) return;
    int src = edges[gwave];
    int dst = edges[E + gwave];

    v4f vs = *(const v4f*)(nodes + (size_t)src * D_DIM + lane * 4);
    v4f vd = *(const v4f*)(nodes + (size_t)dst * D_DIM + lane * 4);

    float* a1 = agg1 + (size_t)dst * D_DIM + lane * 4;
    float* a2 = agg2 + (size_t)src * D_DIM + lane * 4;
    unsafeAtomicAdd(a1 + 0, vs.x);
    unsafeAtomicAdd(a1 + 1, vs.y);
    unsafeAtomicAdd(a1 + 2, vs.z);
    unsafeAtomicAdd(a1 + 3, vs.w);
    unsafeAtomicAdd(a2 + 0, vd.x);
    unsafeAtomicAdd(a2 + 1, vd.y);
    unsafeAtomicAdd(a2 + 2, vd.z);
    unsafeAtomicAdd(a2 + 3, vd.w);
    if (lane == 0) {
        unsafeAtomicAdd(cnt + dst, 1.0f);
        unsafeAtomicAdd(cnt + src, 1.0f);
    }
}

// ---------------------------------------------------------------------------
// Pass 2: scale aggregated rows by 1/cnt (0 if cnt==0).  Row-scaling commutes
// with the later right-multiplication by W^T, so the GEMM needs no per-row fix.
// ---------------------------------------------------------------------------
__global__ void __launch_bounds__(256) rgcn_scale_kernel(
    float* __restrict__ agg1, float* __restrict__ agg2,
    const float* __restrict__ cnt, int N)
{
    int i = blockIdx.x * blockDim.x + threadIdx.x;
    if (i >= N * 32) return;
    int row = i >> 5;
    int off = (i & 31) * 4;
    float c = cnt[row];
    float s = (c > 0.0f) ? (1.0f / c) : 0.0f;
    v4f* p1 = (v4f*)(agg1 + (size_t)row * D_DIM + off);
    v4f* p2 = (v4f*)(agg2 + (size_t)row * D_DIM + off);
    v4f x1 = *p1, x2 = *p2;
    x1 *= s; x2 *= s;
    *p1 = x1; *p2 = x2;
}

// ---------------------------------------------------------------------------
// Pass 3: fused GEMM + ReLU using V_WMMA_F32_16X16X4_F32 (full-precision f32).
// out = relu(nodes@W0^T + agg1s@W1^T + agg2s@W2^T)
// Block = 256 threads = 8 wave32s; each wave owns a 16x128 output strip.
// Each weight matrix is staged into 64KB LDS pre-swizzled into the WMMA
// B-fragment layout (lane[0:15]=N cols, half-wave selects K parity), so the
// inner loop is ds_load_b64 + v_wmma per 16x16 tile per K-step-of-4.
// ---------------------------------------------------------------------------
__global__ void __launch_bounds__(256) rgcn_gemm_relu_kernel(
    const float* __restrict__ nodes, const float* __restrict__ agg1,
    const float* __restrict__ agg2, const float* __restrict__ W0,
    const float* __restrict__ W1, const float* __restrict__ W2,
    float* __restrict__ out, int N)
{
    __shared__ v2f ldsB[8192];   // 64 KB: one 128x128 weight matrix per pass

    const int tid  = threadIdx.x;
    const int lane = tid & 31;
    const int wave = tid >> 5;
    const int half = lane >> 4;     // K-parity half-wave (ISA A/B layout)
    const int n16  = lane & 15;
    const int rowbase = blockIdx.x * 128 + wave * 16;
    const int arow = rowbase + n16;
    const size_t aro = (size_t)((arow < N) ? arow : (N - 1)) * D_DIM;

    v8f acc[8] = {};   // 8 col-tiles x 16x16 f32 accumulator (64 VGPRs)

    for (int m = 0; m < 3; ++m) {
        const float* W = (m == 0) ? W0    : ((m == 1) ? W1   : W2);
        const float* X = (m == 0) ? nodes : ((m == 1) ? agg1 : agg2);

        // Stage W into LDS in B-fragment order:
        //   flat f = t*1024 + ks*32 + lane  ->  W[t*16 + (lane&15)][ks*4 + 2*(lane>>4) + {0,1}]
        for (int f = tid; f < 8192; f += 256) {
            int l2 = f & 31;
            int ks = (f >> 5) & 31;
            int tt = f >> 10;
            int j  = tt * 16 + (l2 & 15);
            int k  = ks * 4 + 2 * (l2 >> 4);
            ldsB[f] = *(const v2f*)(W + j * D_DIM + k);
        }
        __syncthreads();

#pragma unroll 4
        for (int ks = 0; ks < 32; ++ks) {
            // A fragment: lane holds row (rowbase + lane&15), K = ks*4 + 2*half .. +1
            v2f a = *(const v2f*)(X + aro + ks * 4 + 2 * half);
#pragma unroll
            for (int t = 0; t < 8; ++t) {
                v2f b = ldsB[t * 1024 + ks * 32 + lane];
                acc[t] = __builtin_amdgcn_wmma_f32_16x16x4_f32(
                    false, a, false, b, (short)0, acc[t], false, false);
            }
        }
        __syncthreads();
    }

    // Epilogue: ReLU + store.  C/D layout: VGPR g holds row (g + 8*half), lane&15 = col.
#pragma unroll
    for (int t = 0; t < 8; ++t) {
#pragma unroll
        for (int g = 0; g < 8; ++g) {
            int row = rowbase + g + 8 * half;
            if (row < N) {
                out[(size_t)row * D_DIM + t * 16 + n16] = fmaxf(acc[t][g], 0.0f);
            }
        }
    }
}

extern "C" void kernel_launch(void* const* d_in, const int* in_sizes, int n_in,
                              void* d_out, int out_size, void* d_ws, size_t ws_size,
                              hipStream_t stream)
{
    const float* nodes = (const float*)d_in[0];
    const int*   edges = (const int*)d_in[1];
    const float* W0    = (const float*)d_in[2];
    const float* W1    = (const float*)d_in[3];
    const float* W2    = (const float*)d_in[4];
    float* out = (float*)d_out;

    const int N = in_sizes[0] / D_DIM;   // 50000
    const int E = in_sizes[1] / 2;       // 600000

    float* agg1 = (float*)d_ws;
    float* agg2 = agg1 + (size_t)N * D_DIM;
    float* cnt  = agg2 + (size_t)N * D_DIM;
    size_t zero_bytes = ((size_t)2 * N * D_DIM + (size_t)N) * sizeof(float);
    hipMemsetAsync(d_ws, 0, zero_bytes, stream);

    // Pass 1: one wave per edge -> 8 edges per 256-thread block
    int scatterBlocks = (E + 7) / 8;
    rgcn_scatter_kernel<<<scatterBlocks, 256, 0, stream>>>(
        nodes, edges, agg1, agg2, cnt, E);

    // Pass 2: N*32 threads, float4 per thread over both agg buffers
    int scaleThreads = N * 32;
    rgcn_scale_kernel<<<(scaleThreads + 255) / 256, 256, 0, stream>>>(
        agg1, agg2, cnt, N);

    // Pass 3: 128 rows per block
    int gemmBlocks = (N + 127) / 128;
    rgcn_gemm_relu_kernel<<<gemmBlocks, 256, 0, stream>>>(
        nodes, agg1, agg2, W0, W1, W2, out, N);
}